// CoupledTwistorLNN_5660766896348
// MI455X (gfx1250) — compile-verified
//
#include <hip/hip_runtime.h>

typedef __attribute__((ext_vector_type(2))) float v2f;
typedef __attribute__((ext_vector_type(8))) float v8f;

#define T_STEPS 2048
#define BATCH   4096
#define DT_C    0.1f
#define CPL     0.1f

static __device__ __forceinline__ v8f splat8(float s) {
  v8f r;
#pragma unroll
  for (int i = 0; i < 8; ++i) r[i] = s;
  return r;
}

// fast reciprocal: v_rcp_f32 (avoids the IEEE v_div_scale/div_fmas/div_fixup chain)
static __device__ __forceinline__ float frcp(float x) { return __builtin_amdgcn_rcpf(x); }
// raw v_sqrt_f32 (skip libm denormal-rescale fixup)
static __device__ __forceinline__ float fsqrt(float x) { return __builtin_amdgcn_sqrtf(x); }

// scalar tanh: native CDNA5 v_tanh_f32 when the toolchain declares it
static __device__ __forceinline__ float ftanh(float x) {
#if __has_builtin(__builtin_amdgcn_tanhf)
  return __builtin_amdgcn_tanhf(x);
#else
  float e = __expf(2.0f * x);
  return 1.0f - 2.0f * frcp(e + 1.0f);
#endif
}

static __device__ __forceinline__ v8f rcp8(v8f x) {
  v8f r;
#pragma unroll
  for (int i = 0; i < 8; ++i) r[i] = frcp(x[i]);
  return r;
}

static __device__ __forceinline__ v8f sig8(v8f x) {
  v8f r;
#pragma unroll
  for (int i = 0; i < 8; ++i) r[i] = frcp(1.0f + __expf(-x[i]));
  return r;
}

static __device__ __forceinline__ v8f tanh8(v8f x) {
  v8f r;
#pragma unroll
  for (int i = 0; i < 8; ++i) r[i] = ftanh(x[i]);
  return r;
}

// D = A(16x16, via 4 K=4 chunks) x B + C   -- f32 WMMA, full precision
static __device__ __forceinline__ v8f wmma4(const v2f a[4], const v2f b[4], v8f c) {
#pragma unroll
  for (int k = 0; k < 4; ++k)
    c = __builtin_amdgcn_wmma_f32_16x16x4_f32(false, a[k], false, b[k], (short)0, c,
                                              false, false);
  return c;
}
// D = A(16x8, via 2 K=4 chunks) x B + C
static __device__ __forceinline__ v8f wmma2(const v2f a[2], const v2f b[2], v8f c) {
#pragma unroll
  for (int k = 0; k < 2; ++k)
    c = __builtin_amdgcn_wmma_f32_16x16x4_f32(false, a[k], false, b[k], (short)0, c,
                                              false, false);
  return c;
}

// Row-major 16x16 weight W[j][i]  ->  WMMA-B layout blocks: B_kc[k'][n] = W[n][4kc+k']
// B layout (f32 4x16): lane = n + 16*(k'>=2), vgpr = k'%2
static __device__ __forceinline__ void loadB16(const float* __restrict__ W, int n, int half,
                                               v2f b[4]) {
  const float* p = W + n * 16 + 2 * half;
#pragma unroll
  for (int k = 0; k < 4; ++k) b[k] = *(const v2f*)(p + 4 * k);
}
// Row-major 16x8 weight U[j][i] -> B layout (2 K-chunks)
static __device__ __forceinline__ void loadB8(const float* __restrict__ U, int n, int half,
                                              v2f b[2]) {
  const float* p = U + n * 8 + 2 * half;
#pragma unroll
  for (int k = 0; k < 2; ++k) b[k] = *(const v2f*)(p + 4 * k);
}

// C/D layout (f32 16x16: lane = n + 16*(m>=8), vgpr = m%8) -> LDS row-major [m*16+n]
static __device__ __forceinline__ void storeC(float* s, int n, int half, v8f c) {
  const int m0 = half * 8;
#pragma unroll
  for (int v = 0; v < 8; ++v) s[(m0 + v) * 16 + n] = c[v];
}
// LDS row-major -> A layout (f32 16x4 blocks: lane = m + 16*(k'>=2), vgpr = k'%2)
static __device__ __forceinline__ void loadA(const float* s, int row, int half, v2f a[4]) {
  const float* p = s + row * 16 + 2 * half;
#pragma unroll
  for (int k = 0; k < 4; ++k) a[k] = *(const v2f*)(p + 4 * k);
}

__global__ __launch_bounds__(32) void ctlnn_scan_kernel(
    const float* __restrict__ x,
    const float* __restrict__ Wh,   const float* __restrict__ bWh,
    const float* __restrict__ Uh,   const float* __restrict__ bUh,
    const float* __restrict__ Wth,  const float* __restrict__ bWth,
    const float* __restrict__ b_h,
    const float* __restrict__ Wzr,  const float* __restrict__ bWzr,
    const float* __restrict__ Wzi,  const float* __restrict__ bWzi,
    const float* __restrict__ Uz,   const float* __restrict__ bUz,
    const float* __restrict__ Wtz,  const float* __restrict__ bWtz,
    const float* __restrict__ b_zr, const float* __restrict__ b_zi,
    const float* __restrict__ Wh2z, const float* __restrict__ bh2z,
    const float* __restrict__ Wz2h, const float* __restrict__ bz2h,
    const float* __restrict__ Woh,  const float* __restrict__ bOh,
    const float* __restrict__ Woz,  const float* __restrict__ bOz,
    float* __restrict__ y) {
  __shared__ float sH[256];
  __shared__ float sZR[256];
  __shared__ float sZI[256];

  const int lane = threadIdx.x;
  const int half = lane >> 4;
  const int n    = lane & 15;
  const int b0   = blockIdx.x * 16;

  // ---- preload weights into WMMA-B layout registers (persist across all T) ----
  v2f Bwh[4], Bwth[4], Bwzr[4], Bwzi[4], Bwtz[4], Bh2z[4], Bz2h[4];
  loadB16(Wh, n, half, Bwh);
  loadB16(Wth, n, half, Bwth);
  loadB16(Wzr, n, half, Bwzr);
  loadB16(Wzi, n, half, Bwzi);
  loadB16(Wtz, n, half, Bwtz);
  loadB16(Wh2z, n, half, Bh2z);
  loadB16(Wz2h, n, half, Bz2h);
  v2f Buh[2], Buz[2];
  loadB8(Uh, n, half, Buh);
  loadB8(Uz, n, half, Buz);

  // per-lane bias scalars: C/D layout column bias = bias[lane%16] in all 8 vgprs
  const float vbWh  = bWh[n],  vbUh  = bUh[n],  vbWth = bWth[n], vb_h  = b_h[n];
  const float vbWzr = bWzr[n], vbWzi = bWzi[n], vbUz  = bUz[n],  vbWtz = bWtz[n];
  const float vb_zr = b_zr[n], vb_zi = b_zi[n], vbh2z = bh2z[n], vbz2h = bz2h[n];

  // wave-uniform output weights (scalar regs)
  float woh[16], woz[16];
#pragma unroll
  for (int i = 0; i < 16; ++i) { woh[i] = Woh[i]; woz[i] = Woz[i]; }
  const float ybias = bOh[0] + bOz[0];

  // ---- states: C/D layout for elementwise, A layout for next matmul ----
  v8f hC = splat8(0.f), zrC = splat8(0.f), ziC = splat8(0.f);
  v2f aH[4], aZR[4], aZI[4], aZA[4];
#pragma unroll
  for (int k = 0; k < 4; ++k) { aH[k] = 0.f; aZR[k] = 0.f; aZI[k] = 0.f; aZA[k] = 0.f; }

  for (int t = 0; t < T_STEPS; ++t) {
    // x tile in A layout (16 batch x 8 in, two K=4 chunks) straight from global
    const float* px = x + ((size_t)t * BATCH + b0 + n) * 8 + 2 * half;
    v2f ax[2];
    ax[0] = *(const v2f*)(px);
    ax[1] = *(const v2f*)(px + 4);
    if (t + 1 < T_STEPS) __builtin_prefetch(px + BATCH * 8, 0, 1);

    // ---- behavior space ----
    v8f tauh_inv = rcp8(sig8(wmma4(aH, Bwth, splat8(vbWth))) + 1e-6f);
    v8f coup_z   = wmma4(aZR, Bz2h, splat8(vbz2h)) * CPL;
    v8f th       = tanh8(wmma4(aH, Bwh, splat8(vbWh)));
    v8f uhx      = wmma2(ax, Buh, splat8(vbUh));
    v8f dh       = (th - hC + uhx + splat8(vb_h) + coup_z) * tauh_inv;

    // ---- structure space ----
    v8f tauz_inv = rcp8(sig8(wmma4(aZA, Bwtz, splat8(vbWtz))) + 1e-6f);
    v8f coup_h   = wmma4(aH, Bh2z, splat8(vbh2z)) * CPL;
    v8f uzx      = wmma2(ax, Buz, splat8(vbUz));
    v8f tzr      = tanh8(wmma4(aZR, Bwzr, splat8(vbWzr)));
    v8f tzi      = tanh8(wmma4(aZI, Bwzi, splat8(vbWzi)));
    v8f dzr      = (tzr - zrC + uzx + splat8(vb_zr) + coup_h) * tauz_inv;
    v8f dzi      = (tzi - ziC + uzx + splat8(vb_zi) + coup_h) * tauz_inv;

    hC  = hC + dh * DT_C;
    zrC = zrC + dzr * DT_C;
    ziC = ziC + dzi * DT_C;

    // ---- marshal C/D -> LDS -> A layout for next step ----
    storeC(sH, n, half, hC);
    storeC(sZR, n, half, zrC);
    storeC(sZI, n, half, ziC);
    __syncthreads();
    loadA(sH, n, half, aH);
    loadA(sZR, n, half, aZR);
    loadA(sZI, n, half, aZI);
#pragma unroll
    for (int k = 0; k < 4; ++k) {  // |z| is elementwise: compute directly in A layout
      aZA[k].x = fsqrt(aZR[k].x * aZR[k].x + aZI[k].x * aZI[k].x);
      aZA[k].y = fsqrt(aZR[k].y * aZR[k].y + aZI[k].y * aZI[k].y);
    }

    // ---- output: per-lane dot over LDS rows (lanes 0..15 = batch rows) ----
    if (lane < 16) {
      float acc = ybias;
      const float* hr = sH + lane * 16;
      const float* zr = sZR + lane * 16;
#pragma unroll
      for (int i = 0; i < 16; ++i)
        acc = __fmaf_rn(woh[i], hr[i], __fmaf_rn(woz[i], zr[i], acc));
      y[(size_t)t * BATCH + b0 + lane] = acc;
    }
    __syncthreads();
  }
}

extern "C" void kernel_launch(void* const* d_in, const int* in_sizes, int n_in,
                              void* d_out, int out_size, void* d_ws, size_t ws_size,
                              hipStream_t stream) {
  (void)in_sizes; (void)n_in; (void)d_ws; (void)ws_size; (void)out_size;
  const float* x    = (const float*)d_in[0];
  const float* Wh   = (const float*)d_in[1];
  const float* bWh  = (const float*)d_in[2];
  const float* Uh   = (const float*)d_in[3];
  const float* bUh  = (const float*)d_in[4];
  const float* Wth  = (const float*)d_in[5];
  const float* bWth = (const float*)d_in[6];
  const float* b_h  = (const float*)d_in[7];
  const float* Wzr  = (const float*)d_in[8];
  const float* bWzr = (const float*)d_in[9];
  const float* Wzi  = (const float*)d_in[10];
  const float* bWzi = (const float*)d_in[11];
  const float* Uz   = (const float*)d_in[12];
  const float* bUz  = (const float*)d_in[13];
  const float* Wtz  = (const float*)d_in[14];
  const float* bWtz = (const float*)d_in[15];
  const float* b_zr = (const float*)d_in[16];
  const float* b_zi = (const float*)d_in[17];
  const float* Wh2z = (const float*)d_in[18];
  const float* bh2z = (const float*)d_in[19];
  const float* Wz2h = (const float*)d_in[20];
  const float* bz2h = (const float*)d_in[21];
  const float* Woh  = (const float*)d_in[22];
  const float* bOh  = (const float*)d_in[23];
  const float* Woz  = (const float*)d_in[24];
  const float* bOz  = (const float*)d_in[25];
  float* y = (float*)d_out;

  dim3 grid(BATCH / 16);  // 256 waves, one 16-row batch tile each, persistent over T
  dim3 block(32);
  ctlnn_scan_kernel<<<grid, block, 0, stream>>>(
      x, Wh, bWh, Uh, bUh, Wth, bWth, b_h, Wzr, bWzr, Wzi, bWzi, Uz, bUz, Wtz, bWtz,
      b_zr, b_zi, Wh2z, bh2z, Wz2h, bz2h, Woh, bOh, Woz, bOz, y);
}